// FoldingNet_43628277793055
// MI455X (gfx1250) — compile-verified
//
#include <hip/hip_runtime.h>

// ---------------------------------------------------------------------------
// FoldingNet decode, fully fused for gfx1250 (wave32, WMMA f32 16x16x4).
//   B=128, N=2048, LATENT=256.
//   pre1[b][64] = x[b] @ W1[:, :256]^T + b1      (precomputed, WMMA)
//   pre4[b][64] = x[b] @ W4[:, :256]^T + b4      (precomputed, WMMA)
//   main kernel: per 16-point tile per wave, all 6 layers fused in
//   registers/LDS, fp32 WMMA for the 64->16 / 16->16 contractions.
// ---------------------------------------------------------------------------

typedef float v2f __attribute__((ext_vector_type(2)));
typedef float v8f __attribute__((ext_vector_type(8)));

#define EPSF 1e-5f

static __device__ __forceinline__ v8f wmma_f32_k4(v2f a, v2f b, v8f c) {
  // D = A(16x4) * B(4x16) + C(16x16), fp32
  return __builtin_amdgcn_wmma_f32_16x16x4_f32(false, a, false, b, (short)0, c,
                                               false, false);
}

// ---------------------------------------------------------------------------
// Kernel 1: pre1 / pre4 via WMMA.  64 blocks x 32 threads.
//   block id -> { which matrix (W1 / W4), m-tile (batch/16), n-tile (out/16) }
// ---------------------------------------------------------------------------
__global__ __launch_bounds__(32) void foldingnet_pre(
    const float* __restrict__ x,
    const float* __restrict__ W1, const float* __restrict__ b1,
    const float* __restrict__ W4, const float* __restrict__ b4,
    float* __restrict__ pre1, float* __restrict__ pre4) {
  const int lane = threadIdx.x & 31;
  const int id = blockIdx.x;          // 0..63
  const int which = id >> 5;          // 0 -> W1/pre1, 1 -> W4/pre4
  const int tile = id & 31;           // 0..31
  const int mt = tile >> 2;           // batch tile 0..7
  const int nt = tile & 3;            // output tile 0..3

  const float* W    = which ? W4 : W1;
  const float* bias = which ? b4 : b1;
  float*       pre  = which ? pre4 : pre1;
  const int    lead = which ? 259 : 258;

  const int m16 = lane & 15;
  const int half = lane >> 4;
  const int row  = mt * 16 + m16;     // batch index (A row)
  const int ncol = nt * 16 + m16;     // output feature (B column)

  v8f acc = {};
  #pragma unroll
  for (int k0 = 0; k0 < 256; k0 += 4) {
    const int kb = k0 + half * 2;     // A: vgpr0 holds K=kb, vgpr1 K=kb+1
    v2f a;  a.x = x[row * 256 + kb];      a.y = x[row * 256 + kb + 1];
    v2f bb; bb.x = W[ncol * lead + kb];   bb.y = W[ncol * lead + kb + 1];
    acc = wmma_f32_k4(a, bb, acc);
  }
  const float bv = bias[ncol];
  #pragma unroll
  for (int j = 0; j < 8; ++j) {
    const int m = mt * 16 + half * 8 + j;         // D row
    pre[m * 64 + nt * 16 + m16] = acc[j] + bv;    // D col = lane&15
  }
}

// ---------------------------------------------------------------------------
// Kernel 2: fused per-point pipeline.  256 threads = 8 waves/block,
// each wave owns one 16-point tile.  2048 blocks * 8 = 16384 tiles.
// ---------------------------------------------------------------------------
__global__ __launch_bounds__(256) void foldingnet_main(
    const float* __restrict__ W1,
    const float* __restrict__ W2, const float* __restrict__ b2,
    const float* __restrict__ W3, const float* __restrict__ b3,
    const float* __restrict__ W4,
    const float* __restrict__ W5, const float* __restrict__ b5,
    const float* __restrict__ W6, const float* __restrict__ b6,
    const float* __restrict__ g1, const float* __restrict__ be1,
    const float* __restrict__ g2, const float* __restrict__ be2,
    const float* __restrict__ g3, const float* __restrict__ be3,
    const float* __restrict__ g4, const float* __restrict__ be4,
    const float* __restrict__ g5, const float* __restrict__ be5,
    const float* __restrict__ pre1, const float* __restrict__ pre4,
    float* __restrict__ out) {
  // per-wave LDS region: 16x68 (64-wide acts, padded) + 16x17 (16-wide,
  // transposed) + 16x4 (3-wide acts)
  constexpr int WREG = 16 * 68 + 16 * 17 + 16 * 4;   // 1424 floats
  __shared__ float smem[8 * WREG];

  const int lane = threadIdx.x & 31;
  const int wave = threadIdx.x >> 5;
  float* sA = smem + wave * WREG;   // [16][68]
  float* sT = sA + 16 * 68;         // [16][17]
  float* s3 = sT + 16 * 17;         // [16][4]

  const int tileId = blockIdx.x * 8 + wave;   // 0..16383
  const int b = tileId >> 7;                  // batch element
  const int t = tileId & 127;                 // point tile within batch
  const int m16 = lane & 15;
  const int half = lane >> 4;

  // ---- Stage 1: h1 = pre1[b] + grid-part, LN64, ReLU -> sA[m][o] ----------
  {
    const int n = t * 16 + m16;               // global point for lane's row
    const float sq = 45.254833995939045f;     // sqrt(2048)
    const float lin0 = (float)n * (4096.0f / 2047.0f);
    const float gr0 = truncf(lin0 / sq) / sq - 1.0f;
    const float lin1 = (float)n * (2048.0f / 2047.0f);
    const float gr1 = 2.0f * fmodf(lin1, sq) / sq - 1.0f;

    const float* p1 = pre1 + b * 64;
    float hv[32];
    float sum = 0.f, ssq = 0.f;
    #pragma unroll
    for (int i = 0; i < 32; ++i) {
      const int o = half * 32 + i;
      const float h = p1[o] + gr0 * W1[o * 258 + 256] + gr1 * W1[o * 258 + 257];
      hv[i] = h; sum += h; ssq += h * h;
    }
    sum += __shfl_xor(sum, 16, 32);   // combine the two feature halves
    ssq += __shfl_xor(ssq, 16, 32);
    const float mn = sum * (1.f / 64.f);
    const float vr = ssq * (1.f / 64.f) - mn * mn;
    const float rs = rsqrtf(vr + EPSF);
    #pragma unroll
    for (int i = 0; i < 32; ++i) {
      const int o = half * 32 + i;
      const float a = (hv[i] - mn) * rs * g1[o] + be1[o];
      sA[m16 * 68 + o] = a > 0.f ? a : 0.f;
    }
  }

  // ---- Stage 2: h2 = a1 @ W2^T (K=64), +b2, LN16, ReLU -> sT[m][n] --------
  {
    v8f acc = {};
    #pragma unroll
    for (int k0 = 0; k0 < 64; k0 += 4) {
      const int kb = k0 + half * 2;
      v2f a;  a.x = sA[m16 * 68 + kb];   a.y = sA[m16 * 68 + kb + 1];
      v2f bb; bb.x = W2[m16 * 64 + kb];  bb.y = W2[m16 * 64 + kb + 1];
      acc = wmma_f32_k4(a, bb, acc);
    }
    const float bv = b2[m16], gv = g2[m16], ev = be2[m16];
    #pragma unroll
    for (int j = 0; j < 8; ++j) {
      const float v = acc[j] + bv;
      float sj = v, qj = v * v;
      #pragma unroll
      for (int off = 1; off < 16; off <<= 1) {   // reduce over N=16 lanes
        sj += __shfl_xor(sj, off, 32);
        qj += __shfl_xor(qj, off, 32);
      }
      const float mn = sj * (1.f / 16.f);
      const float vr = qj * (1.f / 16.f) - mn * mn;
      const float rs = rsqrtf(vr + EPSF);
      float a = (v - mn) * rs * gv + ev;
      a = a > 0.f ? a : 0.f;
      sT[(half * 8 + j) * 17 + m16] = a;   // transpose into [point][feature]
    }
  }

  // ---- Stage 3: h3 = a2 @ W3^T (K=16, 3 outputs), +b3, LN3, ReLU ----------
  {
    const int nc = m16 < 3 ? m16 : 2;
    const float msk = m16 < 3 ? 1.f : 0.f;   // select, not branch: EXEC full
    v8f acc = {};
    #pragma unroll
    for (int k0 = 0; k0 < 16; k0 += 4) {
      const int kb = k0 + half * 2;
      v2f a;  a.x = sT[m16 * 17 + kb];        a.y = sT[m16 * 17 + kb + 1];
      v2f bb; bb.x = W3[nc * 16 + kb] * msk;  bb.y = W3[nc * 16 + kb + 1] * msk;
      acc = wmma_f32_k4(a, bb, acc);
    }
    const float bv = b3[nc];
    #pragma unroll
    for (int j = 0; j < 8; ++j) {
      if (m16 < 3) s3[(half * 8 + j) * 4 + m16] = acc[j] + bv;
    }
    // per-point LN over 3 features (both halves compute identical values)
    const float v0 = s3[m16 * 4 + 0], v1 = s3[m16 * 4 + 1], v2 = s3[m16 * 4 + 2];
    const float mn = (v0 + v1 + v2) * (1.f / 3.f);
    const float d0 = v0 - mn, d1 = v1 - mn, d2 = v2 - mn;
    const float vr = (d0 * d0 + d1 * d1 + d2 * d2) * (1.f / 3.f);
    const float rs = rsqrtf(vr + EPSF);
    float a0 = d0 * rs * g3[0] + be3[0]; a0 = a0 > 0.f ? a0 : 0.f;
    float a1 = d1 * rs * g3[1] + be3[1]; a1 = a1 > 0.f ? a1 : 0.f;
    float a2 = d2 * rs * g3[2] + be3[2]; a2 = a2 > 0.f ? a2 : 0.f;
    s3[m16 * 4 + 0] = a0; s3[m16 * 4 + 1] = a1; s3[m16 * 4 + 2] = a2;
  }

  // ---- Stage 4: h4 = pre4[b] + a3 @ W4[:,256:259]^T, LN64, ReLU -> sA -----
  {
    const float* p4 = pre4 + b * 64;
    const float a0 = s3[m16 * 4 + 0];
    const float a1 = s3[m16 * 4 + 1];
    const float a2 = s3[m16 * 4 + 2];
    float hv[32];
    float sum = 0.f, ssq = 0.f;
    #pragma unroll
    for (int i = 0; i < 32; ++i) {
      const int o = half * 32 + i;
      const float* wr = W4 + o * 259 + 256;
      const float h = p4[o] + a0 * wr[0] + a1 * wr[1] + a2 * wr[2];
      hv[i] = h; sum += h; ssq += h * h;
    }
    sum += __shfl_xor(sum, 16, 32);
    ssq += __shfl_xor(ssq, 16, 32);
    const float mn = sum * (1.f / 64.f);
    const float vr = ssq * (1.f / 64.f) - mn * mn;
    const float rs = rsqrtf(vr + EPSF);
    #pragma unroll
    for (int i = 0; i < 32; ++i) {
      const int o = half * 32 + i;
      const float a = (hv[i] - mn) * rs * g4[o] + be4[o];
      sA[m16 * 68 + o] = a > 0.f ? a : 0.f;
    }
  }

  // ---- Stage 5: h5 = a4 @ W5^T (K=64), +b5, LN16, ReLU -> sT --------------
  {
    v8f acc = {};
    #pragma unroll
    for (int k0 = 0; k0 < 64; k0 += 4) {
      const int kb = k0 + half * 2;
      v2f a;  a.x = sA[m16 * 68 + kb];   a.y = sA[m16 * 68 + kb + 1];
      v2f bb; bb.x = W5[m16 * 64 + kb];  bb.y = W5[m16 * 64 + kb + 1];
      acc = wmma_f32_k4(a, bb, acc);
    }
    const float bv = b5[m16], gv = g5[m16], ev = be5[m16];
    #pragma unroll
    for (int j = 0; j < 8; ++j) {
      const float v = acc[j] + bv;
      float sj = v, qj = v * v;
      #pragma unroll
      for (int off = 1; off < 16; off <<= 1) {
        sj += __shfl_xor(sj, off, 32);
        qj += __shfl_xor(qj, off, 32);
      }
      const float mn = sj * (1.f / 16.f);
      const float vr = qj * (1.f / 16.f) - mn * mn;
      const float rs = rsqrtf(vr + EPSF);
      float a = (v - mn) * rs * gv + ev;
      a = a > 0.f ? a : 0.f;
      sT[(half * 8 + j) * 17 + m16] = a;
    }
  }

  // ---- Stage 6: out = a5 @ W6^T (K=16, 3 outputs) + b6 --------------------
  {
    const int nc = m16 < 3 ? m16 : 2;
    v8f acc = {};
    #pragma unroll
    for (int k0 = 0; k0 < 16; k0 += 4) {
      const int kb = k0 + half * 2;
      v2f a;  a.x = sT[m16 * 17 + kb];  a.y = sT[m16 * 17 + kb + 1];
      v2f bb; bb.x = W6[nc * 16 + kb];  bb.y = W6[nc * 16 + kb + 1];
      acc = wmma_f32_k4(a, bb, acc);   // columns n>=3 computed but never stored
    }
    const float bv = b6[nc];
    if (m16 < 3) {
      #pragma unroll
      for (int j = 0; j < 8; ++j) {
        const int mm = half * 8 + j;
        const int np = t * 16 + mm;
        out[((long)b * 2048 + np) * 3 + m16] = acc[j] + bv;
      }
    }
  }
}

// ---------------------------------------------------------------------------
extern "C" void kernel_launch(void* const* d_in, const int* in_sizes, int n_in,
                              void* d_out, int out_size, void* d_ws,
                              size_t ws_size, hipStream_t stream) {
  const float* x   = (const float*)d_in[0];
  const float* W1  = (const float*)d_in[1];
  const float* b1  = (const float*)d_in[2];
  const float* W2  = (const float*)d_in[3];
  const float* b2  = (const float*)d_in[4];
  const float* W3  = (const float*)d_in[5];
  const float* b3  = (const float*)d_in[6];
  const float* W4  = (const float*)d_in[7];
  const float* b4  = (const float*)d_in[8];
  const float* W5  = (const float*)d_in[9];
  const float* b5  = (const float*)d_in[10];
  const float* W6  = (const float*)d_in[11];
  const float* b6  = (const float*)d_in[12];
  const float* g1  = (const float*)d_in[13];
  const float* be1 = (const float*)d_in[14];
  const float* g2  = (const float*)d_in[15];
  const float* be2 = (const float*)d_in[16];
  const float* g3  = (const float*)d_in[17];
  const float* be3 = (const float*)d_in[18];
  const float* g4  = (const float*)d_in[19];
  const float* be4 = (const float*)d_in[20];
  const float* g5  = (const float*)d_in[21];
  const float* be5 = (const float*)d_in[22];

  float* pre1 = (float*)d_ws;            // 128*64 floats
  float* pre4 = pre1 + 128 * 64;         // 128*64 floats (64 KiB total)

  foldingnet_pre<<<64, 32, 0, stream>>>(x, W1, b1, W4, b4, pre1, pre4);
  foldingnet_main<<<2048, 256, 0, stream>>>(
      W1, W2, b2, W3, b3, W4, W5, b5, W6, b6,
      g1, be1, g2, be2, g3, be3, g4, be4, g5, be5,
      pre1, pre4, (float*)d_out);
}